// GaussianMomentDescriptorFlax_26920855011349
// MI455X (gfx1250) — compile-verified
//
#include <hip/hip_runtime.h>
#include <hip/hip_bf16.h>

#define N_RADIAL 5
#define N_BASIS  7
#define N_SPEC   119
#define ACC_STRIDE 112   // 100 unique moment floats per atom + zeroed pad
#define OUT_COLS 360
#define WAVES_PER_BLOCK 8

typedef float v2f __attribute__((ext_vector_type(2)));
typedef float v8f __attribute__((ext_vector_type(8)));

// ---------------------------------------------------------------------------
// Compile-time index tables.
// Unified per-wave LDS buffer layout (floats):
//   [0..99]    symmetric-unique moments (m0:0, m1:5, m2u:20, m3u:50)
//   [100..110] guaranteed zero (acc pad, memset-zeroed, never atomically hit)
//   [111]      constant 1.0f (written by lane 0)
//   [112..367] OP  = m1 (x) m1          (16x16)
//   [368..623] G4  = m2 . m2 (i-contr)  (16x16)
//   [624..879] H   = m3 . m3 (ij-contr) (16x16)
//   [880..1135]K   = m3 . m2 (ij-contr) (16x16)
// ---------------------------------------------------------------------------
#define OFF_M1 5
#define OFF_M2 20
#define OFF_M3 50
#define IDX_ZERO 110
#define IDX_ONE  111
#define OP_B 112
#define G_B  368
#define H_B  624
#define K_B  880
#define BUF_FLOATS 1136

constexpr int cP2(int x, int y) {
    int h = x > y ? x : y, l = x > y ? y : x;
    return h * (h + 1) / 2 + l;
}
constexpr int cP3(int x, int y, int z) {
    int a = x > y ? x : y; a = a > z ? a : z;
    int c = x < y ? x : y; c = c < z ? c : z;
    int b = x + y + z - a - c;
    return a * (a + 1) * (a + 2) / 6 + b * (b + 1) / 2 + c;
}
constexpr int pack(int a, int b) { return (a & 0xffff) | (b << 16); }

struct DevTables {
    int aG[32];        // packed (e0,e1) LDS indices for G4 A-fragment
    int aOP[32];       // ... for OP A-fragment
    int aH[3][32];     // ... for H/K A-fragment per K-step
    int bK[3][32];     // ... for K B-fragment per K-step
    int idx[OUT_COLS][9];   // packed (A,B) buffer indices per output term
};

constexpr DevTables make_tables() {
    DevTables t{};
    // tril enumerations used by the reference
    int t2r[15] = {}, t2s[15] = {};
    int t3r[35] = {}, t3s[35] = {}, t3t[35] = {};
    {
        int p = 0;
        for (int r = 0; r < 5; ++r)
            for (int s = 0; s <= r; ++s) { t2r[p] = r; t2s[p] = s; ++p; }
        p = 0;
        for (int r = 0; r < 5; ++r)
            for (int s = 0; s <= r; ++s)
                for (int u = 0; u <= s; ++u) { t3r[p] = r; t3s[p] = s; t3t[p] = u; ++p; }
    }
    // fragment gather tables (A 16x4 f32 layout: lane M=l%16, K=e+2*(l/16))
    for (int l = 0; l < 32; ++l) {
        int M = l & 15, hi = l >> 4;
        int ag[2] = {IDX_ZERO, IDX_ZERO}, ao[2] = {IDX_ZERO, IDX_ZERO};
        for (int e = 0; e < 2; ++e) {
            int K = e + 2 * hi;
            if (M < 15 && K < 3) ag[e] = OFF_M2 + (M / 3) * 6 + cP2(K, M % 3);
            if (M < 15 && K == 0) ao[e] = OFF_M1 + M;
        }
        t.aG[l]  = pack(ag[0], ag[1]);
        t.aOP[l] = pack(ao[0], ao[1]);
        for (int st = 0; st < 3; ++st) {
            int ah[2] = {IDX_ZERO, IDX_ZERO}, bk[2] = {IDX_ZERO, IDX_ZERO};
            for (int e = 0; e < 2; ++e) {
                int q = 4 * st + e + 2 * hi;       // flattened ij 0..8 (+pad)
                if (q < 9) {
                    int qi = q / 3, qj = q % 3;
                    if (M < 15) ah[e] = OFF_M3 + (M / 3) * 10 + cP3(qi, qj, M % 3);
                    if (M < 5)  bk[e] = OFF_M2 + M * 6 + cP2(qi, qj);
                }
            }
            t.aH[st][l] = pack(ah[0], ah[1]);
            t.bK[st][l] = pack(bk[0], bk[1]);
        }
    }
    // output term tables: every column is sum_q buf[A]*buf[B]
    for (int c = 0; c < OUT_COLS; ++c)
        for (int q = 0; q < 9; ++q) t.idx[c][q] = pack(IDX_ZERO, IDX_ZERO);
    for (int c = 0; c < 5; ++c)                           // contr_0 = m0
        t.idx[c][0] = pack(c, IDX_ONE);
    for (int p = 0; p < 15; ++p) {                        // c1,c2,c3 traces
        int r = t2r[p], s = t2s[p];
        for (int q = 0; q < 3; ++q) {
            int d = (3 * r + q) * 16 + (3 * s + q);
            t.idx[5  + p][q] = pack(OP_B + d, IDX_ONE);
            t.idx[20 + p][q] = pack(G_B  + d, IDX_ONE);
            t.idx[35 + p][q] = pack(H_B  + d, IDX_ONE);
        }
    }
    for (int p = 0; p < 35; ++p) {                        // c4 = G4 : m2
        int r = t3r[p], s = t3s[p], u = t3t[p];
        for (int j = 0; j < 3; ++j)
            for (int k = 0; k < 3; ++k)
                t.idx[50 + p][j * 3 + k] =
                    pack(G_B + (3 * r + j) * 16 + 3 * s + k,
                         OFF_M2 + u * 6 + cP2(j, k));
    }
    for (int p = 0; p < 75; ++p) {                        // c5, c6
        int pr = p / 5, u = p % 5;
        int r = t2r[pr], s = t2s[pr];
        for (int i = 0; i < 3; ++i)
            for (int j = 0; j < 3; ++j) {
                int b = OFF_M2 + u * 6 + cP2(i, j);
                t.idx[85  + p][i * 3 + j] =
                    pack(OP_B + (3 * r + i) * 16 + 3 * s + j, b);  // c5
                t.idx[160 + p][i * 3 + j] =
                    pack(H_B  + (3 * r + i) * 16 + 3 * s + j, b);  // c6
            }
    }
    for (int p = 0; p < 125; ++p) {                       // c7 = K . m1
        int r = p / 25, s = (p / 5) % 5, u = p % 5;
        for (int k = 0; k < 3; ++k)
            t.idx[235 + p][k] = pack(K_B + (3 * r + k) * 16 + s,
                                     OFF_M1 + u * 3 + k);
    }
    return t;
}

__constant__ DevTables TBL = make_tables();

// ---------------------------------------------------------------------------
// Phase 1: per-edge moments scatter-added (f32 no-return atomics -> L2)
// ---------------------------------------------------------------------------
__device__ __forceinline__ int rP2(int x, int y) {
    int h = x > y ? x : y, l = x > y ? y : x;
    return h * (h + 1) / 2 + l;
}
__device__ __forceinline__ int rP3(int x, int y, int z) {
    int a = max(x, max(y, z)), c = min(x, min(y, z));
    int b = x + y + z - a - c;
    return a * (a + 1) * (a + 2) / 6 + b * (b + 1) / 2 + c;
}

__global__ __launch_bounds__(256)
void gm_edge_kernel(const float* __restrict__ R, const int* __restrict__ Z,
                    const int* __restrict__ nbr, const float* __restrict__ W,
                    float* __restrict__ acc, int n_edges)
{
    int e = blockIdx.x * blockDim.x + threadIdx.x;
    if (e >= n_edges) return;

    int ai = nbr[e];
    int aj = nbr[n_edges + e];
    if (ai == aj) return;

    float dx = R[3 * aj + 0] - R[3 * ai + 0];
    float dy = R[3 * aj + 1] - R[3 * ai + 1];
    float dz = R[3 * aj + 2] - R[3 * ai + 2];
    float dr = sqrtf(dx * dx + dy * dy + dz * dz);
    if (dr >= 6.0f) return;                        // cutoff == 0

    float inv = 1.0f / (dr + 1e-5f);
    float dn[3] = { dx * inv, dy * inv, dz * inv };

    const float betta    = 49.0f / 36.0f;
    const float rad_norm = 0.9648141f;             // (2*betta/pi)^0.25
    const float embed    = 0.37796447f;            // 1/sqrt(7)
    float basis[N_BASIS];
#pragma unroll
    for (int n = 0; n < N_BASIS; ++n) {
        float d = dr - (0.5f + (5.5f / 7.0f) * (float)n);
        basis[n] = rad_norm * __expf(-betta * d * d);
    }
    float cut = 0.5f * (__cosf((float)M_PI * dr * (1.0f / 6.0f)) + 1.0f);

    int zi = Z[ai], zj = Z[aj];
    const float* w = W + (size_t)(zi * N_SPEC + zj) * (N_RADIAL * N_BASIS);

    float rad[N_RADIAL];
#pragma unroll
    for (int r = 0; r < N_RADIAL; ++r) {
        float s = 0.0f;
#pragma unroll
        for (int n = 0; n < N_BASIS; ++n) s += basis[n] * w[r * N_BASIS + n];
        rad[r] = s * cut * embed;
    }

    float dnp2[6], dnp3[10];
#pragma unroll
    for (int i = 0; i < 3; ++i)
#pragma unroll
        for (int j = 0; j <= i; ++j) dnp2[rP2(i, j)] = dn[i] * dn[j];
#pragma unroll
    for (int i = 0; i < 3; ++i)
#pragma unroll
        for (int j = 0; j <= i; ++j)
#pragma unroll
            for (int k = 0; k <= j; ++k) dnp3[rP3(i, j, k)] = dn[i] * dn[j] * dn[k];

    float* a = acc + (size_t)aj * ACC_STRIDE;
#pragma unroll
    for (int r = 0; r < N_RADIAL; ++r) {
        float rr = rad[r];
        unsafeAtomicAdd(a + r, rr);                               // m0
#pragma unroll
        for (int i = 0; i < 3; ++i)
            unsafeAtomicAdd(a + OFF_M1 + r * 3 + i, rr * dn[i]);  // m1
#pragma unroll
        for (int p = 0; p < 6; ++p)
            unsafeAtomicAdd(a + OFF_M2 + r * 6 + p, rr * dnp2[p]);
#pragma unroll
        for (int p = 0; p < 10; ++p)
            unsafeAtomicAdd(a + OFF_M3 + r * 10 + p, rr * dnp3[p]);
    }
}

// ---------------------------------------------------------------------------
// Phase 2: one wave32 per atom; 8 x V_WMMA_F32_16X16X4_F32.
// Gram trick: f32 B-fragment of A^T has identical lane contents to the
// A-fragment, so Gram/outer products pass the same registers twice.
// ---------------------------------------------------------------------------
__device__ __forceinline__ v8f wmma4(v2f a, v2f b, v8f c) {
    return __builtin_amdgcn_wmma_f32_16x16x4_f32(false, a, false, b,
                                                 (short)0, c, false, false);
}
__device__ __forceinline__ v2f frag(const float* s, int pk) {
    v2f f;
    f[0] = s[pk & 0xffff];
    f[1] = s[((unsigned)pk) >> 16];
    return f;
}

__global__ __launch_bounds__(256)
void gm_atom_kernel(const float* __restrict__ acc, float* __restrict__ out,
                    int n_atoms)
{
    __shared__ float buf[WAVES_PER_BLOCK][BUF_FLOATS];

    const int wave = threadIdx.x >> 5;
    const int lane = threadIdx.x & 31;
    const int Mrow = lane & 15;
    const int hi   = lane >> 4;

    int atom  = blockIdx.x * WAVES_PER_BLOCK + wave;
    int atomc = atom < n_atoms ? atom : (n_atoms - 1);   // keep EXEC full

    float* bw = buf[wave];
    const float* g = acc + (size_t)atomc * ACC_STRIDE;
    for (int t = lane; t < ACC_STRIDE; t += 32) bw[t] = g[t];  // pads are 0
    if (lane == 0) bw[IDX_ONE] = 1.0f;
    __syncthreads();

    // OP = m1 (x) m1  (rank-1, K=0 only) and G4 = m2 . m2 (K = i)
    v8f zero = {0.f, 0.f, 0.f, 0.f, 0.f, 0.f, 0.f, 0.f};
    v2f aO = frag(bw, TBL.aOP[lane]);
    v2f aG = frag(bw, TBL.aG[lane]);
    v8f dO = wmma4(aO, aO, zero);
    v8f dG = wmma4(aG, aG, zero);

    // H = m3 . m3 and K = m3 . m2 over flattened ij (K=9, 3 chained steps)
    v8f dH = zero, dK = zero;
#pragma unroll
    for (int st = 0; st < 3; ++st) {
        v2f aH = frag(bw, TBL.aH[st][lane]);
        v2f bK = frag(bw, TBL.bK[st][lane]);
        dH = wmma4(aH, aH, dH);
        dK = wmma4(aH, bK, dK);
    }

    // spill D fragments: D[M = v + 8*hi, N = lane%16]
#pragma unroll
    for (int v = 0; v < 8; ++v) {
        int row = (v + 8 * hi) * 16 + Mrow;
        bw[OP_B + row] = dO[v];
        bw[G_B  + row] = dG[v];
        bw[H_B  + row] = dH[v];
        bw[K_B  + row] = dK[v];
    }
    __syncthreads();

    if (atom >= n_atoms) return;

    float* o = out + (size_t)atom * OUT_COLS;
    for (int col = lane; col < OUT_COLS; col += 32) {
        float v = 0.0f;
#pragma unroll
        for (int q = 0; q < 9; ++q) {
            int pk = TBL.idx[col][q];
            v += bw[pk & 0xffff] * bw[((unsigned)pk) >> 16];
        }
        o[col] = v;
    }
}

// ---------------------------------------------------------------------------
extern "C" void kernel_launch(void* const* d_in, const int* in_sizes, int n_in,
                              void* d_out, int out_size, void* d_ws, size_t ws_size,
                              hipStream_t stream)
{
    const float* R   = (const float*)d_in[0];
    const int*   Z   = (const int*)  d_in[1];
    const int*   nbr = (const int*)  d_in[2];
    const float* W   = (const float*)d_in[4];
    float* out = (float*)d_out;
    float* acc = (float*)d_ws;

    int n_atoms = in_sizes[1];
    int n_edges = in_sizes[2] / 2;

    hipMemsetAsync(acc, 0, (size_t)n_atoms * ACC_STRIDE * sizeof(float), stream);

    int eb = (n_edges + 255) / 256;
    gm_edge_kernel<<<eb, 256, 0, stream>>>(R, Z, nbr, W, acc, n_edges);

    int ab = (n_atoms + WAVES_PER_BLOCK - 1) / WAVES_PER_BLOCK;
    gm_atom_kernel<<<ab, 256, 0, stream>>>(acc, out, n_atoms);
}